// SelfAttentionGraphNetwork_78237124264286
// MI455X (gfx1250) — compile-verified
//
#include <hip/hip_runtime.h>
#include <math.h>

#define NN 50000
#define EE 800000
#define DD 128
#define KK1 25000
#define KK2 12500
#define BN_EPS 1e-5f

typedef __attribute__((ext_vector_type(2))) float v2f;
typedef __attribute__((ext_vector_type(8))) float v8f;

// ---------------- edge init / aggregation ----------------

__global__ void edge_init_kernel(const int* __restrict__ ei, int* __restrict__ src,
                                 int* __restrict__ dst, int nE) {
  int e = blockIdx.x * blockDim.x + threadIdx.x;
  if (e < nE) { src[e] = ei[e]; dst[e] = ei[nE + e]; }
}

// one wave (32 lanes) per edge; lanes stripe the 128-float feature row
__global__ void edge_scatter_kernel(const float* __restrict__ x,
                                    const int* __restrict__ src,
                                    const int* __restrict__ dst,
                                    float* __restrict__ agg,
                                    int* __restrict__ deg,
                                    int n, int nE) {
  int wid = (blockIdx.x * blockDim.x + threadIdx.x) >> 5;
  int lane = threadIdx.x & 31;
  if (wid >= nE) return;
  int s = src[wid], d = dst[wid];
  // self loops and dummy edges (src==dst==k) land in the dropped segment
  if (s == d || (unsigned)d >= (unsigned)n || (unsigned)s >= (unsigned)n) return;
  const float* xs = x + (size_t)s * DD;
  float* ad = agg + (size_t)d * DD;
#pragma unroll
  for (int r = 0; r < 4; ++r) atomicAdd(&ad[lane + 32 * r], xs[lane + 32 * r]);
  if (lane == 0) atomicAdd(&deg[d], 1);
}

// ---------------- SAGE linear via f32 WMMA (16x16x4) ----------------
// out = relu( l2norm_row( ((agg+x)/(deg+1)) @ W + b ) )
__global__ void sage_gemm_kernel(const float* __restrict__ agg,
                                 const float* __restrict__ x,
                                 const int* __restrict__ deg,
                                 const float* __restrict__ W,
                                 const float* __restrict__ bias,
                                 float* __restrict__ xout, int n) {
  __shared__ float ldsA[16 * DD];
  __shared__ float ldsO[16 * DD];
  __shared__ float ldsN[16];
  const int tid = threadIdx.x;
  const int m0 = blockIdx.x * 16;

  // stage A tile = (agg + x) / (deg + 1) into LDS
#pragma unroll
  for (int t = 0; t < 8; ++t) {
    int idx = tid + t * 256;
    int r = idx >> 7, c = idx & 127;
    int gi = m0 + r;
    float v = 0.f;
    if (gi < n) {
      size_t o = (size_t)gi * DD + c;
      v = (agg[o] + x[o]) / ((float)deg[gi] + 1.0f);
    }
    ldsA[idx] = v;
  }
  __syncthreads();

  const int wave = tid >> 5;        // 8 waves -> 8 N-tiles of 16
  const int lane = tid & 31;
  const int n0 = wave * 16;
  const int lm = lane & 15;
  const int lh = lane >> 4;

  v8f acc = {0.f, 0.f, 0.f, 0.f, 0.f, 0.f, 0.f, 0.f};
  for (int k0 = 0; k0 < DD; k0 += 4) {
    v2f a, b;
    // A 16x4 f32 layout: lane%16 = M ; VGPR v -> K = v + 2*(lane/16)
    a.x = ldsA[lm * DD + k0 + 0 + 2 * lh];
    a.y = ldsA[lm * DD + k0 + 1 + 2 * lh];
    // B 4x16 mirrored layout: lane%16 = N ; VGPR v -> K = v + 2*(lane/16)
    b.x = W[(size_t)(k0 + 0 + 2 * lh) * DD + n0 + lm];
    b.y = W[(size_t)(k0 + 1 + 2 * lh) * DD + n0 + lm];
    acc = __builtin_amdgcn_wmma_f32_16x16x4_f32(false, a, false, b, (short)0,
                                                acc, false, false);
  }

  // C/D layout: VGPR v -> M = v + 8*(lane/16) ; N = n0 + lane%16
#pragma unroll
  for (int v = 0; v < 8; ++v) {
    int m = v + 8 * lh;
    int col = n0 + lm;
    ldsO[m * DD + col] = acc[v] + bias[col];
  }
  __syncthreads();

  if (tid < 16) {
    float s = 0.f;
    for (int j = 0; j < DD; ++j) { float t = ldsO[tid * DD + j]; s += t * t; }
    ldsN[tid] = fmaxf(sqrtf(s), 1e-12f);
  }
  __syncthreads();

#pragma unroll
  for (int t = 0; t < 8; ++t) {
    int idx = tid + t * 256;
    int r = idx >> 7, c = idx & 127;
    int gi = m0 + r;
    if (gi < n) xout[(size_t)gi * DD + c] = fmaxf(ldsO[idx] / ldsN[r], 0.f);
  }
}

// ---------------- BatchNorm (deterministic two-pass) ----------------

__global__ void bn_partial_kernel(const float* __restrict__ x, int n,
                                  float* __restrict__ pS, float* __restrict__ pQ) {
  int j = threadIdx.x;  // 128
  float s = 0.f, q = 0.f;
  for (int i = blockIdx.x; i < n; i += gridDim.x) {
    float v = x[(size_t)i * DD + j];
    s += v; q += v * v;
  }
  pS[blockIdx.x * DD + j] = s;
  pQ[blockIdx.x * DD + j] = q;
}

__global__ void bn_finalize_kernel(const float* __restrict__ pS,
                                   const float* __restrict__ pQ, int P, int n,
                                   float* __restrict__ mv) {
  int j = threadIdx.x;
  float s = 0.f, q = 0.f;
  for (int p = 0; p < P; ++p) { s += pS[p * DD + j]; q += pQ[p * DD + j]; }
  float m = s / (float)n;
  mv[j] = m;
  mv[DD + j] = fmaxf(q / (float)n - m * m, 0.f);  // biased variance
}

__global__ void bn_apply_kernel(float* __restrict__ x, const float* __restrict__ mv,
                                const float* __restrict__ g,
                                const float* __restrict__ b, int n) {
  int idx = blockIdx.x * blockDim.x + threadIdx.x;
  if (idx >= n * DD) return;
  int j = idx & 127;
  x[idx] = (x[idx] - mv[j]) * rsqrtf(mv[DD + j] + BN_EPS) * g[j] + b[j];
}

// ---------------- column max (order-independent atomic) ----------------

__global__ void colmax_kernel(const float* __restrict__ x, int n,
                              unsigned* __restrict__ enc) {
  int idx = blockIdx.x * blockDim.x + threadIdx.x;
  if (idx >= n * DD) return;
  int j = idx & 127;
  int b = __float_as_int(x[idx]);
  unsigned key = (b >= 0) ? ((unsigned)b + 0x80000000u) : ~((unsigned)b);
  atomicMax(&enc[j], key);
}

// ---------------- SAGPooling ----------------

__global__ void pool_score_kernel(const float* __restrict__ agg,
                                  const float* __restrict__ x,
                                  const int* __restrict__ deg,
                                  const float* __restrict__ wp,
                                  const float* __restrict__ bpool, int sidx,
                                  float* __restrict__ score) {
  __shared__ float red[DD];
  int i = blockIdx.x;
  int j = threadIdx.x;
  size_t o = (size_t)i * DD + j;
  float a = (agg[o] + x[o]) / ((float)deg[i] + 1.0f);
  red[j] = a * wp[j];
  __syncthreads();
  for (int s = 64; s > 0; s >>= 1) {
    if (j < s) red[j] += red[j + s];
    __syncthreads();
  }
  if (j == 0) score[i] = tanhf(red[0] + bpool[sidx]);
}

// rank = #(score_j > score_i) + #(score_j == score_i && j < i)  (stable desc)
__global__ void topk_rank_kernel(const float* __restrict__ score, int n,
                                 int* __restrict__ rank) {
  __shared__ float tile[256];
  int i = blockIdx.x * 256 + threadIdx.x;
  float my = (i < n) ? score[i] : 0.f;
  int cnt = 0;
  int nt = (n + 255) >> 8;
  for (int t = 0; t < nt; ++t) {
    int jj = t * 256 + threadIdx.x;
    tile[threadIdx.x] = (jj < n) ? score[jj] : -2.f;  // tanh in [-1,1]
    __syncthreads();
    if (i < n) {
      int base = t * 256;
      int lim = min(256, n - base);
      for (int u = 0; u < lim; ++u) {
        float s = tile[u];
        cnt += (int)((s > my) || (s == my && (base + u) < i));
      }
    }
    __syncthreads();
  }
  if (i == n) rank[i] = 0x7fffffff;   // dummy node never selected
  else if (i < n) rank[i] = cnt;
}

__global__ void pool_compact_kernel(const float* __restrict__ x,
                                    const float* __restrict__ score,
                                    const int* __restrict__ rank, int n, int k,
                                    float* __restrict__ x2) {
  int idx = blockIdx.x * blockDim.x + threadIdx.x;
  if (idx >= n * DD) return;
  int i = idx >> 7, j = idx & 127;
  int r = rank[i];
  if (r < k) x2[(size_t)r * DD + j] = x[idx] * score[i];
}

__global__ void edge_remap_kernel(int* __restrict__ src, int* __restrict__ dst,
                                  const int* __restrict__ rank, int k, int nE) {
  int e = blockIdx.x * blockDim.x + threadIdx.x;
  if (e >= nE) return;
  int rs = rank[src[e]];
  int rd = rank[dst[e]];
  bool valid = (rs < k) && (rd < k);
  src[e] = valid ? rs : k;
  dst[e] = valid ? rd : k;
}

// ---------------- head MLP ----------------

__global__ void head_kernel(const unsigned* __restrict__ enc,
                            const float* __restrict__ W1, const float* __restrict__ b1,
                            const float* __restrict__ W2, const float* __restrict__ b2,
                            float* __restrict__ out) {
  __shared__ float outs[3 * DD];
  __shared__ float h[50];
  int t = threadIdx.x;  // 128
  for (int r = 0; r < 3; ++r) {
    unsigned key = enc[r * DD + t];
    unsigned bits = (key >= 0x80000000u) ? (key - 0x80000000u) : ~key;
    outs[r * DD + t] = __int_as_float((int)bits);
  }
  __syncthreads();
  if (t < 50) {
    float s = b1[t];
    for (int kx = 0; kx < 3 * DD; ++kx) s += outs[kx] * W1[kx * 50 + t];
    h[t] = fmaxf(s, 0.f);
  }
  __syncthreads();
  if (t < 5) {
    float s = b2[t];
    for (int c = 0; c < 50; ++c) s += h[c] * W2[c * 5 + t];
    out[t] = s;
  }
}

// ---------------- orchestration ----------------

extern "C" void kernel_launch(void* const* d_in, const int* in_sizes, int n_in,
                              void* d_out, int out_size, void* d_ws, size_t ws_size,
                              hipStream_t stream) {
  (void)in_sizes; (void)n_in; (void)out_size; (void)ws_size;

  const float* x0    = (const float*)d_in[0];
  const int*   ei    = (const int*)d_in[1];
  const float* Wc    = (const float*)d_in[2];
  const float* bc    = (const float*)d_in[3];
  const float* gamma = (const float*)d_in[4];
  const float* beta  = (const float*)d_in[5];
  const float* Wpool = (const float*)d_in[6];
  const float* bpool = (const float*)d_in[7];
  const float* W1    = (const float*)d_in[8];
  const float* b1    = (const float*)d_in[9];
  const float* W2    = (const float*)d_in[10];
  const float* b2    = (const float*)d_in[11];

  char* ws = (char*)d_ws;
  size_t off = 0;
  auto alloc = [&](size_t bytes) -> void* {
    void* p = ws + off;
    off = (off + bytes + 255) & ~(size_t)255;
    return p;
  };

  float*    xA   = (float*)alloc((size_t)NN * DD * 4);
  float*    xB   = (float*)alloc((size_t)NN * DD * 4);
  float*    agg  = (float*)alloc((size_t)(NN + 1) * DD * 4);
  int*      deg  = (int*)alloc((size_t)(NN + 1) * 4);
  float*    scr  = (float*)alloc((size_t)NN * 4);
  int*      rnk  = (int*)alloc((size_t)(NN + 1) * 4);
  int*      srcB = (int*)alloc((size_t)EE * 4);
  int*      dstB = (int*)alloc((size_t)EE * 4);
  const int P = 256;
  float*    pS   = (float*)alloc((size_t)P * DD * 4);
  float*    pQ   = (float*)alloc((size_t)P * DD * 4);
  float*    mv   = (float*)alloc(2 * DD * 4);
  unsigned* enc  = (unsigned*)alloc(3 * DD * 4);

  const int bn_idx[9] = {0, 1, 2, 3, 4, 5, 6, 6, 8};  // replicate source bug
  const int ks[2] = {KK1, KK2};

  hipMemsetAsync(enc, 0, 3 * DD * 4, stream);
  edge_init_kernel<<<(EE + 255) / 256, 256, 0, stream>>>(ei, srcB, dstB, EE);

  const float* cur = x0;
  int n = NN;
  for (int s = 0; s < 3; ++s) {
    for (int i = 0; i < 3; ++i) {
      int li = 3 * s + i;
      hipMemsetAsync(agg, 0, (size_t)(n + 1) * DD * 4, stream);
      hipMemsetAsync(deg, 0, (size_t)(n + 1) * 4, stream);
      edge_scatter_kernel<<<(EE + 7) / 8, 256, 0, stream>>>(cur, srcB, dstB, agg,
                                                            deg, n, EE);
      float* outb = (cur == xA) ? xB : xA;
      sage_gemm_kernel<<<(n + 15) / 16, 256, 0, stream>>>(
          agg, cur, deg, Wc + (size_t)li * DD * DD, bc + li * DD, outb, n);
      bn_partial_kernel<<<P, DD, 0, stream>>>(outb, n, pS, pQ);
      bn_finalize_kernel<<<1, DD, 0, stream>>>(pS, pQ, P, n, mv);
      bn_apply_kernel<<<(n * DD + 255) / 256, 256, 0, stream>>>(
          outb, mv, gamma + bn_idx[li] * DD, beta + bn_idx[li] * DD, n);
      cur = outb;
    }
    colmax_kernel<<<(n * DD + 255) / 256, 256, 0, stream>>>(cur, n, enc + s * DD);
    if (s < 2) {
      int k = ks[s];
      hipMemsetAsync(agg, 0, (size_t)(n + 1) * DD * 4, stream);
      hipMemsetAsync(deg, 0, (size_t)(n + 1) * 4, stream);
      edge_scatter_kernel<<<(EE + 7) / 8, 256, 0, stream>>>(cur, srcB, dstB, agg,
                                                            deg, n, EE);
      pool_score_kernel<<<n, DD, 0, stream>>>(agg, cur, deg, Wpool + s * DD,
                                              bpool, s, scr);
      topk_rank_kernel<<<(n + 1 + 255) / 256, 256, 0, stream>>>(scr, n, rnk);
      float* outb = (cur == xA) ? xB : xA;
      pool_compact_kernel<<<(n * DD + 255) / 256, 256, 0, stream>>>(cur, scr, rnk,
                                                                    n, k, outb);
      edge_remap_kernel<<<(EE + 255) / 256, 256, 0, stream>>>(srcB, dstB, rnk, k, EE);
      cur = outb;
      n = k;
    }
  }
  head_kernel<<<1, DD, 0, stream>>>(enc, W1, b1, W2, b2, (float*)d_out);
}